// MeanPool_15290083574043
// MI455X (gfx1250) — compile-verified
//
#include <hip/hip_runtime.h>
#include <stdint.h>

// MeanPool over variable-length bags: one workgroup (512 thr = 16 waves) per bag.
// - Stream H rows in 4-row x 512-col chunks via global_load_async_to_lds_b128
//   (ASYNCcnt-tracked, 4-deep pipeline).
// - Steady-state loop uses a constant s_wait_asynccnt 3; drain phase does one
//   s_wait_asynccnt 0 + one barrier, then computes the last <=3 chunks with no
//   further waits/barriers (buffers are never rewritten after the last issue).
// - Sum rows with V_WMMA_F32_16X16X4_F32: A = ones(16x4), B = 4x16 tile of H;
//   accumulator carries running column sums (all 16 output rows identical).
// - Tail (<4 rows) added with VALU, then divide by count and store.

typedef __attribute__((ext_vector_type(2))) float v2f;
typedef __attribute__((ext_vector_type(8))) float v8f;

#define DIMD        512
#define CHUNK_ROWS  4
#define CHUNK_ELTS  (CHUNK_ROWS * DIMD)   // 2048 floats = 8 KB
#define CHUNK_BYTES (CHUNK_ELTS * 4)      // 0x2000
#define PIPE        4                     // async pipeline depth (4 LDS buffers)

// Low 32 bits of a generic pointer to __shared__ == LDS byte offset
// (flat LDS aperture keeps the addrspace(3) offset in addr[31:0]).
__device__ __forceinline__ unsigned lds_off32(const void* p) {
  return (unsigned)(uintptr_t)p;
}

// Thread tid moves 4 consecutive floats of the chunk (one async B128 issue
// per thread -> exactly one ASYNCcnt increment per wave per chunk).
// ldsThreadBase = LDS offset of this thread's 16-byte slot in buffer 0;
// slot selects the pipeline buffer via a single shift (no table lookup).
__device__ __forceinline__ void issue_chunk(const float* __restrict__ H,
                                            int rowStart, unsigned ldsThreadBase,
                                            int slot, int tid) {
  int flat = tid * 4;                 // element index within chunk
  int row  = flat >> 9;               // /512
  int col  = flat & (DIMD - 1);
  int goff = ((rowStart + row) * DIMD + col) * 4;     // byte offset < 2^31
  unsigned laddr = ldsThreadBase + ((unsigned)slot * CHUNK_BYTES);
  asm volatile("global_load_async_to_lds_b128 %0, %1, %2"
               :: "v"(laddr), "v"(goff), "s"(H)
               : "memory");
}

__global__ __launch_bounds__(512) void
meanpool_wmma_kernel(const float* __restrict__ H,
                     const int* __restrict__ bag_ptr,
                     float* __restrict__ out) {
  __shared__ float lds[PIPE][CHUNK_ELTS];   // 4 x 8 KB = 32 KB

  const int tid  = threadIdx.x;
  const int wave = tid >> 5;
  const int lane = tid & 31;
  const int bag  = blockIdx.x;

  const int start = bag_ptr[bag];
  const int end   = bag_ptr[bag + 1];
  const int count = end - start;
  const int nchunks = count >> 2;           // full 4-row chunks

  // wave w owns columns [32w, 32w+32): two 16-wide WMMA tiles
  const int c0 = wave * 32 + (lane & 15);
  const int c1 = c0 + 16;
  const int kb = (lane >> 4) * 2;           // B frag K rows: lanes<16 -> K0/K1, else K2/K3

  v2f ones; ones.x = 1.0f; ones.y = 1.0f;   // A = all-ones 16x4 (layout-insensitive)
  v8f acc0 = {}; v8f acc1 = {};

  const unsigned ldsThreadBase = lds_off32(&lds[0][0]) + (unsigned)tid * 16u;

  // 2 DS ops + 2 WMMAs per chunk per wave (ds_load_2addr_stride64 fusion).
  auto compute = [&](int i) {
    const float* buf = &lds[i & (PIPE - 1)][0];
    v2f b0, b1;
    b0.x = buf[kb * DIMD + c0];
    b0.y = buf[(kb + 1) * DIMD + c0];
    b1.x = buf[kb * DIMD + c1];
    b1.y = buf[(kb + 1) * DIMD + c1];
    acc0 = __builtin_amdgcn_wmma_f32_16x16x4_f32(false, ones, false, b0,
                                                 (short)0, acc0, false, false);
    acc1 = __builtin_amdgcn_wmma_f32_16x16x4_f32(false, ones, false, b1,
                                                 (short)0, acc1, false, false);
  };

  if (nchunks > 0) {
    // Prime the pipeline.
    const int pre = nchunks < PIPE ? nchunks : PIPE;
    for (int j = 0; j < pre; ++j)
      issue_chunk(H, start + j * CHUNK_ROWS, ldsThreadBase, j, tid);

    // Steady state: pipeline provably full -> constant wait immediate.
    const int steady = nchunks - (PIPE - 1);   // may be <= 0
    int i = 0;
    for (; i < steady; ++i) {
      asm volatile("s_wait_asynccnt 3" ::: "memory");  // chunk i landed (this wave)
      __syncthreads();                                 // chunk i visible block-wide
      compute(i);
      __syncthreads();                                 // buffer reads done
      if (i + PIPE < nchunks)                          // refill drained slot
        issue_chunk(H, start + (i + PIPE) * CHUNK_ROWS, ldsThreadBase,
                    i & (PIPE - 1), tid);
    }

    // Drain: everything already issued; after one full wait + barrier, all
    // remaining chunks are resident and buffers are never rewritten.
    asm volatile("s_wait_asynccnt 0" ::: "memory");
    __syncthreads();
    for (; i < nchunks; ++i)
      compute(i);
  }

  // Every lane holds the sum for column (lane&15) of its tile in acc[0]
  // (C/D layout: VGPR0 = rows M=0/M=8, all rows identical since A is ones).
  float s0 = acc0[0];
  float s1 = acc1[0];

  // Tail rows (count % 4) via plain global loads.
  for (int r = start + nchunks * CHUNK_ROWS; r < end; ++r) {
    s0 += H[(long)r * DIMD + c0];
    s1 += H[(long)r * DIMD + c1];
  }

  const float cnt = (float)(count > 0 ? count : 1);
  if (lane < 16) {                          // lanes 16-31 hold duplicates
    float* ob = out + (long)bag * DIMD;
    ob[c0] = s0 / cnt;
    ob[c1] = s1 / cnt;
  }
}

extern "C" void kernel_launch(void* const* d_in, const int* in_sizes, int n_in,
                              void* d_out, int out_size, void* d_ws, size_t ws_size,
                              hipStream_t stream) {
  const float* H       = (const float*)d_in[0];
  const int*   bag_ptr = (const int*)d_in[1];
  float*       out     = (float*)d_out;
  const int    nbags   = in_sizes[1] - 1;   // bag_ptr has B+1 entries

  meanpool_wmma_kernel<<<nbags, 512, 0, stream>>>(H, bag_ptr, out);
}